// LSTMModel_29197187678385
// MI455X (gfx1250) — compile-verified
//
#include <hip/hip_runtime.h>

typedef __attribute__((ext_vector_type(2))) float v2f;
typedef __attribute__((ext_vector_type(8))) float v8f;

#define Hh 50
#define Ff 32
#define T_STEPS 365
#define G4 200      // 4*H
#define NPAD 208    // 13 * 16
#define NT 13       // N tiles of 16
#define KH 52       // recurrent K padded to multiple of 4
#define KTX 8       // input-proj K tiles (32/4)
#define KTH 13      // recurrent K tiles (52/4)
#define WAVES 4     // waves cooperating on one 16-row group
#define BLK (WAVES*32)

// LDS layout (float offsets)
#define OFF_WX   0                        // wxT[n][k]  n<208, k<32
#define OFF_WH   (OFF_WX + NPAD*Ff)       // whT[n][k]  n<208, k<52
#define OFF_BIAS (OFF_WH + NPAD*KH)       // bias[208]
#define OFF_H    (OFF_BIAS + NPAD)        // h[16][52]
#define OFF_G    (OFF_H + 16*KH)          // g[16][208]
#define LDS_FLOATS (OFF_G + 16*NPAD)      // 21840 floats = 87360 bytes

__global__ __launch_bounds__(BLK) void lstm_persist_wmma(
    const float* __restrict__ x,
    const float* __restrict__ kernel,
    const float* __restrict__ rec_kernel,
    const float* __restrict__ bias,
    const float* __restrict__ dense_w,
    const float* __restrict__ dense_b,
    float* __restrict__ out)
{
  extern __shared__ float lds[];
  float* wxT = lds + OFF_WX;    // transposed input kernel  [n][k]
  float* whT = lds + OFF_WH;    // transposed recurrent kernel [n][k]
  float* bL  = lds + OFF_BIAS;
  float* hB  = lds + OFF_H;     // h[row][52]  (cols 50,51 stay 0)
  float* gB  = lds + OFF_G;     // g[row][208]

  const int tid   = threadIdx.x;
  const int lane  = tid & 31;
  const int wave  = tid >> 5;
  const int rowA  = lane & 15;
  const int khalf = lane >> 4;           // 0 or 1 (half-wave K split)
  const int ncol  = lane & 15;
  const int b0    = blockIdx.x * 16;

  // N-tile ownership: waves get 3/3/3/4 of the 13 tiles
  const int ntS = (wave * NT) / WAVES;
  const int ntE = ((wave + 1) * NT) / WAVES;
  // elementwise ownership: wave owns gate-units [wave*200, wave*200+200)
  const int ubase = wave * 200;

  // ---- one-time: load weights transposed + zero-padded into LDS ----
  for (int i = tid; i < NPAD * Ff; i += BLK) {
    int n = i / Ff, k = i % Ff;
    wxT[i] = (n < G4) ? kernel[k * G4 + n] : 0.0f;
  }
  for (int i = tid; i < NPAD * KH; i += BLK) {
    int n = i / KH, k = i % KH;
    whT[i] = (n < G4 && k < Hh) ? rec_kernel[k * G4 + n] : 0.0f;
  }
  for (int i = tid; i < NPAD; i += BLK) bL[i] = (i < G4) ? bias[i] : 0.0f;
  for (int i = tid; i < 16 * KH; i += BLK) hB[i] = 0.0f;
  __syncthreads();

  // per-lane cell state: up to 7 gate-units per lane (q*32+lane < 200)
  float c[7];
  #pragma unroll
  for (int q = 0; q < 7; ++q) c[q] = 0.0f;

  const float* xrow = x + (long)(b0 + rowA) * T_STEPS * Ff;

  for (int t = 0; t < T_STEPS; ++t) {
    // A fragments, input projection: row rowA, K = kt*4 + 2*khalf + {0,1}
    const float* xp = xrow + t * Ff;
    v2f ax[KTX];
    #pragma unroll
    for (int kt = 0; kt < KTX; ++kt)
      ax[kt] = *(const v2f*)(xp + kt * 4 + 2 * khalf);
    if (wave == 0 && t + 1 < T_STEPS) __builtin_prefetch(xp + Ff, 0, 3);

    // A fragments, recurrent: h from LDS (padded cols are zero)
    v2f ah[KTH];
    #pragma unroll
    for (int kt = 0; kt < KTH; ++kt)
      ah[kt] = *(const v2f*)(hB + rowA * KH + kt * 4 + 2 * khalf);

    // this wave's share of the 13 N-tiles of 16 gate columns
    for (int nt = ntS; nt < ntE; ++nt) {
      const int n = nt * 16 + ncol;
      v8f acc = {};
      #pragma unroll
      for (int kt = 0; kt < KTX; ++kt) {
        v2f b = *(const v2f*)(wxT + n * Ff + kt * 4 + 2 * khalf);
        acc = __builtin_amdgcn_wmma_f32_16x16x4_f32(
            false, ax[kt], false, b, (short)0, acc, false, false);
      }
      #pragma unroll
      for (int kt = 0; kt < KTH; ++kt) {
        v2f b = *(const v2f*)(whT + n * KH + kt * 4 + 2 * khalf);
        acc = __builtin_amdgcn_wmma_f32_16x16x4_f32(
            false, ah[kt], false, b, (short)0, acc, false, false);
      }
      // C/D layout: VGPR r -> row r + 8*khalf, col = n
      #pragma unroll
      for (int r = 0; r < 8; ++r)
        gB[(r + 8 * khalf) * NPAD + n] = acc[r];
    }
    __syncthreads();   // g complete before gate consumption

    // elementwise gates + state update: 800 units over 128 lanes
    #pragma unroll
    for (int q = 0; q < 7; ++q) {
      int u = q * 32 + lane;
      if (u < 200) {
        int p = ubase + u;
        int row = p / 50, j = p % 50;
        const float* g = gB + row * NPAD;
        float gi = g[j]       + bL[j];
        float gf = g[j +  50] + bL[j +  50];
        float gc = g[j + 100] + bL[j + 100];
        float go = g[j + 150] + bL[j + 150];
        float i_   = 1.0f / (1.0f + __expf(-gi));
        float f_   = 1.0f / (1.0f + __expf(-gf));
        float cand = fmaxf(gc, 0.0f);
        float o_   = 1.0f / (1.0f + __expf(-go));
        float cn = f_ * c[q] + i_ * cand;
        c[q] = cn;
        hB[row * KH + j] = o_ * fmaxf(cn, 0.0f);
      }
    }
    __syncthreads();   // h complete before next step's A_h loads
  }

  // final dense head: out[b] = h . dense_w + dense_b
  if (tid < 16) {
    float s = 0.0f;
    #pragma unroll 10
    for (int j = 0; j < Hh; ++j) s += hB[tid * KH + j] * dense_w[j];
    out[b0 + tid] = s + dense_b[0];
  }
}

extern "C" void kernel_launch(void* const* d_in, const int* in_sizes, int n_in,
                              void* d_out, int out_size, void* d_ws, size_t ws_size,
                              hipStream_t stream) {
  const float* x      = (const float*)d_in[0];
  const float* kernel = (const float*)d_in[1];
  const float* rec    = (const float*)d_in[2];
  const float* bias   = (const float*)d_in[3];
  const float* dw     = (const float*)d_in[4];
  const float* db     = (const float*)d_in[5];
  float* out = (float*)d_out;

  dim3 grid(4096 / 16);   // one workgroup (4 waves) per 16 batch rows
  dim3 block(BLK);
  size_t shmem = (size_t)LDS_FLOATS * sizeof(float);
  hipLaunchKernelGGL(lstm_persist_wmma, grid, block, shmem, stream,
                     x, kernel, rec, bias, dw, db, out);
}